// RayPointRefiner_83150566850731
// MI455X (gfx1250) — compile-verified
//
#include <hip/hip_runtime.h>
#include <cstdint>

#define N_RAYS   524288
#define N_PTS    64
#define N_FINE   64
#define N_OUT    128
#define EPS_F    1e-5f
#define NT       64      // rays (threads) per block; == N_PTS, multiple of wave32

typedef float v4f __attribute__((ext_vector_type(4)));   // native vector for NT store

// float2-column-major LDS index: element j of local ray r.
//  pair jp=j>>1 of ray r lives at float offset 2*(jp*NT + r)  (8B-aligned pairs).
//  Bank for lane r: (2r + (j&1)) mod 64 -> conflict-free for uniform j AND
//  for fully divergent per-lane j (merge pointers).
#define IDX(j, r) ((((j) & ~1) * NT) + (((r)) << 1) + ((j) & 1))

// One thread == one ray.
//  Phase 1: async b64 global->LDS scatter copies stage z & w transposed (ASYNCcnt path).
//  Phase 2: sequential CDF-interval sweep emits the 64 fine samples (no searchsorted).
//  Phase 3: O(128) merge of two sorted lists replaces the sort; packed non-temporal
//           b128 stores (streaming output must not churn the 192 MB L2).
__global__ __launch_bounds__(NT, 1) void ray_point_refine_kernel(
    const float* __restrict__ lengths,
    const float* __restrict__ weights,
    float* __restrict__ out)
{
    __shared__ float zb[N_PTS * NT];   // z values       (16 KB)
    __shared__ float wb[N_PTS * NT];   // raw weights    (16 KB)
    __shared__ float sb[N_FINE * NT];  // fine samples   (16 KB)

    const int t = threadIdx.x;
    const int rayBase = blockIdx.x * NT;

    // ---------------- Phase 1: async b64 staging (CDNA5 ASYNCcnt path) -------------
    // Block tile is contiguous in global: pair index p = it*NT + t covers
    // (local ray r = p>>5, pair jp = p&31).  Coalesced 8B global reads,
    // transposed per-lane LDS scatter writes.
    {
        const uint64_t zg = (uint64_t)(uintptr_t)(lengths + (size_t)rayBase * N_PTS);
        const uint64_t wg = (uint64_t)(uintptr_t)(weights + (size_t)rayBase * N_PTS);
        const uint32_t zb_base = (uint32_t)(uintptr_t)&zb[0]; // low 32 bits = LDS offset
        const uint32_t wb_base = (uint32_t)(uintptr_t)&wb[0];
        #pragma unroll 4
        for (int it = 0; it < N_PTS / 2; ++it) {
            uint32_t p    = (uint32_t)(it * NT + t);     // global float2 index in tile
            uint32_t goff = p * 8u;                      // coalesced across lanes
            uint32_t r    = p >> 5;                      // 32 pairs per ray
            uint32_t jp   = p & 31u;
            uint32_t loff = (jp * NT + r) * 8u;          // transposed (scatter), 8B aligned
            uint32_t zl = zb_base + loff;
            uint32_t wl = wb_base + loff;
            asm volatile("global_load_async_to_lds_b64 %0, %1, %2"
                         :: "v"(zl), "v"(goff), "s"(zg) : "memory");
            asm volatile("global_load_async_to_lds_b64 %0, %1, %2"
                         :: "v"(wl), "v"(goff), "s"(wg) : "memory");
        }
    }
    asm volatile("s_wait_asynccnt 0x0" ::: "memory");
    __syncthreads();

    // ---------------- Phase 2: PDF normalize + interval sweep sampling -------------
    // weights used: w[1..62] (62 bins); bins m[j] = 0.5*(z[j]+z[j+1]), j = 0..62.
    float wsum = 0.0f;
    #pragma unroll
    for (int m = 0; m < N_PTS / 2; ++m) {               // pairwise b64 LDS reads
        float2 v = *(const float2*)&wb[2 * (m * NT + t)];
        float lo = (m == 0)              ? 0.0f : (v.x + EPS_F);   // skip w[0]
        float hi = (m == N_PTS / 2 - 1)  ? 0.0f : (v.y + EPS_F);   // skip w[63]
        wsum += lo + hi;
    }
    const float inv = 1.0f / wsum;

    // cdf: c[0]=0, c[j]=c[j-1]+pdf[j], j=1..62.  searchsorted(cdf,u,'right') for
    // u in [c[j-1], c[j]) gives idx=j -> below=j-1, above=min(j,62).  u_k increasing,
    // so a single walk over intervals j=1..63 (j=63 == "past the end", idx=63).
    const float du = 1.0f / 63.0f;
    int k = 0;
    float run  = 0.0f;               // c[j-1]
    float zjm1 = zb[IDX(0, t)];      // z[j-1]
    float zj   = zb[IDX(1, t)];      // z[j]
    for (int j = 1; j <= 63; ++j) {
        const bool inRange = (j <= 62);
        float zjp1 = inRange ? zb[IDX(j + 1, t)] : zj;
        float mb = 0.5f * (zjm1 + zj);                     // bins[below] = m[j-1]
        float cj, ca, ma;
        if (inRange) {
            cj = run + (wb[IDX(j, t)] + EPS_F) * inv;      // c[j]
            ca = cj;                                       // cdf[above], above=j
            ma = 0.5f * (zj + zjp1);                       // bins[above] = m[j]
        } else {
            cj = 3.0e38f;                                  // +inf sentinel: take the rest
            ca = run;                                      // above==below==62
            ma = mb;
        }
        float cb = run;
        float denom = ca - cb;
        if (denom < EPS_F) denom = 1.0f;
        float rd = 1.0f / denom;                           // hoisted: one div per interval
        float dm = ma - mb;
        while (k < N_FINE) {
            float u = (k == 63) ? 1.0f : (float)k * du;
            if (!(u < cj)) break;                          // side='right' boundary
            sb[IDX(k, t)] = mb + ((u - cb) * rd) * dm;
            ++k;
        }
        if (inRange) run = cj;
        zjm1 = zj; zj = zjp1;
    }

    // ---------------- Phase 3: merge two sorted 64-lists -> 128 out ----------------
    v4f* op = (v4f*)(out + (size_t)(rayBase + t) * N_OUT);   // 512B-aligned per ray
    int i = 0, j2 = 0;
    float a = zb[IDX(0, t)];
    float b = sb[IDX(0, t)];
    #pragma unroll 2
    for (int k2 = 0; k2 < N_OUT / 4; ++k2) {
        v4f v;
        #pragma unroll
        for (int q = 0; q < 4; ++q) {
            bool takeA = (j2 >= N_FINE) || ((i < N_PTS) && (a <= b));
            v[q] = takeA ? a : b;
            if (takeA) {
                ++i;
                a = (i < N_PTS) ? zb[IDX(i, t)] : 3.0e38f;
            } else {
                ++j2;
                b = (j2 < N_FINE) ? sb[IDX(j2, t)] : 3.0e38f;
            }
        }
        // streaming write-once output: non-temporal b128 store (TH=NT)
        __builtin_nontemporal_store(v, op + k2);
    }
}

extern "C" void kernel_launch(void* const* d_in, const int* in_sizes, int n_in,
                              void* d_out, int out_size, void* d_ws, size_t ws_size,
                              hipStream_t stream) {
    // setup_inputs order: origins(0), directions(1), lengths(2), xys(3), ray_weights(4)
    // origins/directions/xys are unused by the reference output.
    const float* lengths  = (const float*)d_in[2];
    const float* rweights = (const float*)d_in[4];
    float* out = (float*)d_out;
    (void)in_sizes; (void)n_in; (void)d_ws; (void)ws_size; (void)out_size;

    dim3 grid(N_RAYS / NT);
    dim3 block(NT);
    ray_point_refine_kernel<<<grid, block, 0, stream>>>(lengths, rweights, out);
}